// SSIMLoss_12910671692213
// MI455X (gfx1250) — compile-verified
//
#include <hip/hip_runtime.h>
#include <hip/hip_bf16.h>

typedef __attribute__((ext_vector_type(2))) float v2f;
typedef __attribute__((ext_vector_type(8))) float v8f;

#define LDSW 66              // padded LDS row stride (floats) to avoid bank conflicts
#define TILE_ELEMS 4096      // 64*64 per (b,c) tile
#define ROW_FLOATS 12288     // 64*64*3 floats per batch tile

// 11-tap Gaussian (sigma=1.5), normalized.
static __device__ const float GW[11] = {
    0.00102838f, 0.00759874f, 0.03600077f, 0.10936069f, 0.21300553f,
    0.26601172f,
    0.21300553f, 0.10936069f, 0.03600077f, 0.00759874f, 0.00102838f
};

// Order-preserving float <-> unsigned key (for integer min/max atomics; deterministic)
__device__ __forceinline__ unsigned fkey(float f) {
    int i = __float_as_int(f);
    return (i >= 0) ? ((unsigned)i | 0x80000000u) : ~(unsigned)i;
}
__device__ __forceinline__ float unkey(unsigned k) {
    int i = (k & 0x80000000u) ? (int)(k ^ 0x80000000u) : (int)~k;
    return __int_as_float(i);
}

__global__ void ssim_init_kernel(unsigned* keys) {
    keys[0] = 0u;           // running max key
    keys[1] = 0xFFFFFFFFu;  // running min key
}

__global__ __launch_bounds__(256)
void ssim_minmax_kernel(const float4* __restrict__ p, unsigned* keys, int n4) {
    float mx = -3.4e38f, mn = 3.4e38f;
    for (int i = blockIdx.x * 256 + threadIdx.x; i < n4; i += gridDim.x * 256) {
        float4 v = p[i];
        mx = fmaxf(mx, fmaxf(fmaxf(v.x, v.y), fmaxf(v.z, v.w)));
        mn = fminf(mn, fminf(fminf(v.x, v.y), fminf(v.z, v.w)));
    }
    for (int off = 16; off > 0; off >>= 1) {
        mx = fmaxf(mx, __shfl_down(mx, off));
        mn = fminf(mn, __shfl_down(mn, off));
    }
    __shared__ float smx[8], smn[8];
    int lane = threadIdx.x & 31, wv = threadIdx.x >> 5;
    if (lane == 0) { smx[wv] = mx; smn[wv] = mn; }
    __syncthreads();
    if (threadIdx.x == 0) {
        for (int i = 1; i < 8; ++i) { mx = fmaxf(mx, smx[i]); mn = fminf(mn, smn[i]); }
        atomicMax(&keys[0], fkey(mx));
        atomicMin(&keys[1], fkey(mn));
    }
}

// One workgroup (4 waves / 128 threads) per (batch, channel) 64x64 tile.
// Separable Gaussian conv as band-matmul: Out = A^T * X * A, WMMA f32 16x16x4,
// skipping K-tiles outside the 11-wide band.
__global__ __launch_bounds__(128)
void ssim_tile_kernel(const float* __restrict__ pred, const float* __restrict__ gt,
                      const unsigned* __restrict__ keys, float* __restrict__ partial) {
    __shared__ float sA[64 * LDSW];      // band matrix A[k][j] = g[k-j], j<54
    __shared__ float sX1[64 * LDSW];
    __shared__ float sX2[64 * LDSW];
    __shared__ float sT[5][64 * LDSW];   // T for x, y, x^2, y^2, xy
    __shared__ float swsum[4];

    const int wg = blockIdx.x;
    const int b  = wg / 3;
    const int c  = wg % 3;
    const int t  = threadIdx.x;

    // Dynamic range logic (matches jnp.where on max/min of img1)
    const float maxf = unkey(keys[0]);
    const float minf = unkey(keys[1]);
    const float maxv = (maxf > 128.0f) ? 255.0f : 1.0f;
    const float minv = (minf < -0.5f) ? -1.0f : 0.0f;
    const float Lr   = maxv - minv;
    const float C1   = (0.01f * Lr) * (0.01f * Lr);
    const float C2   = (0.03f * Lr) * (0.03f * Lr);

    // Build band matrix once per block (cold path; LUT with clamped index).
    for (int idx = t; idx < TILE_ELEMS; idx += 128) {
        int k = idx >> 6, j = idx & 63;
        int d = k - j;
        int dc = d < 0 ? 0 : (d > 10 ? 10 : d);
        float v = GW[dc];
        bool ok = (d >= 0) && (d <= 10) && (j < 54);
        sA[k * LDSW + j] = ok ? v : 0.0f;
    }

    // Load the (b,c) tile: flat = b*12288 + (h*64+w)*3 + c  (stride-3 gather)
    const long base = (long)b * ROW_FLOATS + c;
    for (int idx = t; idx < TILE_ELEMS; idx += 128) {
        int h = idx >> 6, w = idx & 63;
        sX1[h * LDSW + w] = pred[base + (long)idx * 3];
        sX2[h * LDSW + w] = gt[base + (long)idx * 3];
    }
    __syncthreads();

    const int wave = t >> 5;
    const int lane = t & 31;
    const int lm   = lane & 15;   // M (A-frag) / N (B,C,D frags)
    const int hi   = lane >> 4;   // lane-half selector

    // ---------------- Pass 1: T_m = X_m * A  (wave owns row strip m0..m0+15) ----
    {
        const int m0 = wave * 16;
        for (int jt = 0; jt < 4; ++jt) {
            const int j0 = jt * 16;
            v8f a0c = {}, a1c = {}, a2c = {}, a3c = {}, a4c = {};
            const int ktlo = j0 >> 2;
            const int kthi_ = (j0 + 25) >> 2;
            const int kthi = (kthi_ > 15) ? 15 : kthi_;  // band: k in [j0, j0+25]
            for (int kt = ktlo; kt <= kthi; ++kt) {
                const int k0 = kt * 4;
                // A-frag (16x4 f32): lane lm holds (m0+lm, k0+2*hi + {0,1})
                const int ka = k0 + 2 * hi;
                const int rbase = (m0 + lm) * LDSW;
                float x1a = sX1[rbase + ka], x1b = sX1[rbase + ka + 1];
                float x2a = sX2[rbase + ka], x2b = sX2[rbase + ka + 1];
                v2f fa0 = {x1a, x1b};
                v2f fa1 = {x2a, x2b};
                v2f fa2 = {x1a * x1a, x1b * x1b};
                v2f fa3 = {x2a * x2a, x2b * x2b};
                v2f fa4 = {x1a * x2a, x1b * x2b};
                // B-frag (4x16 band): vgpr0 -> K=ka, vgpr1 -> K=ka+1, N=j0+lm
                v2f fb = {sA[ka * LDSW + j0 + lm], sA[(ka + 1) * LDSW + j0 + lm]};
                a0c = __builtin_amdgcn_wmma_f32_16x16x4_f32(false, fa0, false, fb, (short)0, a0c, false, false);
                a1c = __builtin_amdgcn_wmma_f32_16x16x4_f32(false, fa1, false, fb, (short)0, a1c, false, false);
                a2c = __builtin_amdgcn_wmma_f32_16x16x4_f32(false, fa2, false, fb, (short)0, a2c, false, false);
                a3c = __builtin_amdgcn_wmma_f32_16x16x4_f32(false, fa3, false, fb, (short)0, a3c, false, false);
                a4c = __builtin_amdgcn_wmma_f32_16x16x4_f32(false, fa4, false, fb, (short)0, a4c, false, false);
            }
            // C/D layout: vgpr v -> row m0 + v + 8*hi, col j0 + lm
            #pragma unroll
            for (int v = 0; v < 8; ++v) {
                const int off = (m0 + v + 8 * hi) * LDSW + j0 + lm;
                sT[0][off] = a0c[v];
                sT[1][off] = a1c[v];
                sT[2][off] = a2c[v];
                sT[3][off] = a3c[v];
                sT[4][off] = a4c[v];
            }
        }
    }
    __syncthreads();

    // ---------------- Pass 2: Out = A^T * T  (wave owns col strip j0..j0+15) ----
    float lsum = 0.0f;
    {
        const int j0 = wave * 16;
        for (int it = 0; it < 4; ++it) {
            const int i0 = it * 16;
            v8f m1c = {}, m2c = {}, e11c = {}, e22c = {}, e12c = {};
            const int ktlo = i0 >> 2;
            const int kthi_ = (i0 + 25) >> 2;
            const int kthi = (kthi_ > 15) ? 15 : kthi_;  // band: k in [i0, i0+25]
            for (int kt = ktlo; kt <= kthi; ++kt) {
                const int k0 = kt * 4;
                const int ka = k0 + 2 * hi;
                // A-frag = A^T rows: element (i0+lm, ka+{0,1}) = A[k][i]  (same LDS table)
                v2f fa = {sA[ka * LDSW + i0 + lm], sA[(ka + 1) * LDSW + i0 + lm]};
                // B-frags = T rows k, col j0+lm
                const int r0 = ka * LDSW + j0 + lm;
                const int r1 = r0 + LDSW;
                v2f fb0 = {sT[0][r0], sT[0][r1]};
                v2f fb1 = {sT[1][r0], sT[1][r1]};
                v2f fb2 = {sT[2][r0], sT[2][r1]};
                v2f fb3 = {sT[3][r0], sT[3][r1]};
                v2f fb4 = {sT[4][r0], sT[4][r1]};
                m1c  = __builtin_amdgcn_wmma_f32_16x16x4_f32(false, fa, false, fb0, (short)0, m1c,  false, false);
                m2c  = __builtin_amdgcn_wmma_f32_16x16x4_f32(false, fa, false, fb1, (short)0, m2c,  false, false);
                e11c = __builtin_amdgcn_wmma_f32_16x16x4_f32(false, fa, false, fb2, (short)0, e11c, false, false);
                e22c = __builtin_amdgcn_wmma_f32_16x16x4_f32(false, fa, false, fb3, (short)0, e22c, false, false);
                e12c = __builtin_amdgcn_wmma_f32_16x16x4_f32(false, fa, false, fb4, (short)0, e12c, false, false);
            }
            // Elementwise SSIM map, branchless 54x54 mask (keeps EXEC all-ones)
            #pragma unroll
            for (int v = 0; v < 8; ++v) {
                const int i = i0 + v + 8 * hi;
                const int j = j0 + lm;
                float mu1 = m1c[v], mu2 = m2c[v];
                float mu1s = mu1 * mu1, mu2s = mu2 * mu2, m12 = mu1 * mu2;
                float s1  = e11c[v] - mu1s;
                float s2  = e22c[v] - mu2s;
                float s12 = e12c[v] - m12;
                float num = (2.0f * m12 + C1) * (2.0f * s12 + C2);
                float den = (mu1s + mu2s + C1) * (s1 + s2 + C2);
                float val = num / den;  // out-of-range cells: den >= C1*C2 > 0, no NaN
                lsum += ((i < 54) && (j < 54)) ? val : 0.0f;
            }
        }
    }

    // Deterministic block reduction (fixed order)
    for (int off = 16; off > 0; off >>= 1) lsum += __shfl_down(lsum, off);
    if (lane == 0) swsum[wave] = lsum;
    __syncthreads();
    if (t == 0) partial[wg] = (swsum[0] + swsum[1]) + (swsum[2] + swsum[3]);
}

__global__ __launch_bounds__(256)
void ssim_finalize_kernel(const float* __restrict__ partial, int ntile, float* __restrict__ out) {
    __shared__ float red[256];
    float s = 0.0f;
    for (int i = threadIdx.x; i < ntile; i += 256) s += partial[i];  // fixed order
    red[threadIdx.x] = s;
    __syncthreads();
    for (int w = 128; w > 0; w >>= 1) {
        if (threadIdx.x < w) red[threadIdx.x] += red[threadIdx.x + w];
        __syncthreads();
    }
    if (threadIdx.x == 0) {
        const float nvalid = (float)ntile * 54.0f * 54.0f;  // 2048*3*54*54
        out[0] = 1.0f - red[0] / nvalid;
    }
}

extern "C" void kernel_launch(void* const* d_in, const int* in_sizes, int n_in,
                              void* d_out, int out_size, void* d_ws, size_t ws_size,
                              hipStream_t stream) {
    const float* pred = (const float*)d_in[0];
    const float* gt   = (const float*)d_in[1];
    float* out        = (float*)d_out;
    unsigned* keys    = (unsigned*)d_ws;
    float* partial    = (float*)d_ws + 2;

    const int n     = in_sizes[0];        // 2048*4096*3 = 25,165,824
    const int ntile = n / TILE_ELEMS;     // 6144 (b,c) tiles

    ssim_init_kernel<<<1, 1, 0, stream>>>(keys);
    ssim_minmax_kernel<<<1024, 256, 0, stream>>>((const float4*)pred, keys, n / 4);
    ssim_tile_kernel<<<ntile, 128, 0, stream>>>(pred, gt, keys, partial);
    ssim_finalize_kernel<<<1, 256, 0, stream>>>(partial, ntile, out);
}